// ParallelVisualAttentionModule_38886633898232
// MI455X (gfx1250) — compile-verified
//
#include <hip/hip_runtime.h>
#include <hip/hip_bf16.h>
#include <math.h>

// ---------------------------------------------------------------------------
// ParallelVisualAttention for MI455X (gfx1250, wave32).
//   BS=32, L=512, D=512, N_CHARS=32, DICT=97
// GEMMs use V_WMMA_F32_16X16X4_F32 (full f32 precision; problem is far from
// the matrix-throughput roofline, so no need to downcast). The tanh-score
// reduction is inherently VALU (nonlinearity inside the contraction) and is
// done with coalesced loads + wave32 shuffle reductions.
// ---------------------------------------------------------------------------

typedef __attribute__((ext_vector_type(2))) float v2f;
typedef __attribute__((ext_vector_type(8))) float v8f;

#define BSZ   32
#define LSEQ  512
#define DDIM  512
#define NCH   32
#define NDICT 97

#define WMMA_F32(a, b, c) \
  __builtin_amdgcn_wmma_f32_16x16x4_f32(false, (a), false, (b), (short)0, (c), false, false)

// A fragment (16x4 f32): lane L holds A[row=L&15][k0 + (L>>4)*2 + v], v=0,1
__device__ __forceinline__ v2f ldA(const float* __restrict__ A, int lda,
                                   int row_base, int k0, int lane) {
  const int row = row_base + (lane & 15);
  const int k   = k0 + ((lane >> 4) << 1);
  const float* p = A + (size_t)row * lda + k;
  v2f a; a.x = p[0]; a.y = p[1];
  return a;
}

// B fragment (4x16 f32): lane L holds B[k0 + (L>>4)*2 + v][col=L&15], v=0,1
__device__ __forceinline__ v2f ldB(const float* __restrict__ B, int ldb,
                                   int k0, int col_base, int lane) {
  const int col = col_base + (lane & 15);
  const int k   = k0 + ((lane >> 4) << 1);
  v2f b; b.x = B[(size_t)k * ldb + col]; b.y = B[(size_t)(k + 1) * ldb + col];
  return b;
}

// ---------------------------------------------------------------------------
// Kernel 1: w_o = emb(32x512) @ Wo_w(512x512) + Wo_b   (tiny, plain VALU)
// ---------------------------------------------------------------------------
__global__ void wo_gemm_kernel(const float* __restrict__ emb,
                               const float* __restrict__ Wo_w,
                               const float* __restrict__ Wo_b,
                               float* __restrict__ wo) {
  const int idx = blockIdx.x * blockDim.x + threadIdx.x;   // 32*512 threads
  if (idx >= NCH * DDIM) return;
  const int n = idx / DDIM;
  const int c = idx % DDIM;
  float acc = Wo_b[c];
  const float* er = emb + (size_t)n * DDIM;
  for (int k = 0; k < DDIM; ++k)
    acc = fmaf(er[k], Wo_w[(size_t)k * DDIM + c], acc);
  wo[idx] = acc;
}

// ---------------------------------------------------------------------------
// Kernel 2: w_v = inp(16384x512) @ Wv_w(512x512) + Wv_b  via f32 WMMA.
// One wave computes a 16x64 strip (4 C tiles share the A fragment).
// waves = (16384/16) * (512/64) = 8192 ; 8 waves/block -> 1024 blocks.
// ---------------------------------------------------------------------------
__global__ void wv_gemm_wmma_kernel(const float* __restrict__ inp,
                                    const float* __restrict__ Wv_w,
                                    const float* __restrict__ Wv_b,
                                    float* __restrict__ wv) {
  const int lane = threadIdx.x & 31;
  const int wave = (blockIdx.x * blockDim.x + threadIdx.x) >> 5;
  const int NSTRIP = DDIM / 64;                 // 8
  const int row_base = (wave / NSTRIP) * 16;    // 0..16368
  const int col_base = (wave % NSTRIP) * 64;    // 0..448

  v8f c0 = {}, c1 = {}, c2 = {}, c3 = {};
  for (int k0 = 0; k0 < DDIM; k0 += 4) {
    const v2f a  = ldA(inp, DDIM, row_base, k0, lane);
    const v2f b0 = ldB(Wv_w, DDIM, k0, col_base +  0, lane);
    const v2f b1 = ldB(Wv_w, DDIM, k0, col_base + 16, lane);
    const v2f b2 = ldB(Wv_w, DDIM, k0, col_base + 32, lane);
    const v2f b3 = ldB(Wv_w, DDIM, k0, col_base + 48, lane);
    c0 = WMMA_F32(a, b0, c0);
    c1 = WMMA_F32(a, b1, c1);
    c2 = WMMA_F32(a, b2, c2);
    c3 = WMMA_F32(a, b3, c3);
  }

  const int col = col_base + (lane & 15);
  const float bb0 = Wv_b[col], bb1 = Wv_b[col + 16];
  const float bb2 = Wv_b[col + 32], bb3 = Wv_b[col + 48];
  const int rsel = (lane >> 4) << 3;            // 0 or 8
  for (int v = 0; v < 8; ++v) {
    const int row = row_base + v + rsel;
    float* o = wv + (size_t)row * DDIM + col;
    o[0]  = c0[v] + bb0;
    o[16] = c1[v] + bb1;
    o[32] = c2[v] + bb2;
    o[48] = c3[v] + bb3;
  }
}

// ---------------------------------------------------------------------------
// Kernel 3: per-(b,n) workgroup (1024 blocks x 256 threads = 8 waves):
//   e[l]  = sum_d tanh(w_o[n,d] + w_v[b,l,d]) * We_w[d]   (+We_b: softmax-inv.)
//   alpha = softmax_L(e)  -> ws_alpha[(b*NCH+n)*L + l]
// ---------------------------------------------------------------------------
__global__ void attn_softmax_kernel(const float* __restrict__ wo,
                                    const float* __restrict__ wv,
                                    const float* __restrict__ We_w,
                                    float* __restrict__ alpha_out) {
  __shared__ float wo_lds[DDIM];
  __shared__ float we_lds[DDIM];
  __shared__ float e_lds[LSEQ];
  __shared__ float red[256];

  const int b = blockIdx.x / NCH;
  const int n = blockIdx.x % NCH;
  const int tid  = threadIdx.x;
  const int lane = tid & 31;
  const int wid  = tid >> 5;                     // 0..7

  for (int i = tid; i < DDIM; i += 256) {
    wo_lds[i] = wo[(size_t)n * DDIM + i];
    we_lds[i] = We_w[i];                          // We_w is (D,1)
  }
  __syncthreads();

  // --- scores: each wave owns l = wid, wid+8, ... ; lanes stride over d ---
  const float* wv_b = wv + (size_t)b * LSEQ * DDIM;
  for (int l = wid; l < LSEQ; l += 8) {
    const float* pv = wv_b + (size_t)l * DDIM;
    if (l + 8 < LSEQ)
      __builtin_prefetch(wv_b + (size_t)(l + 8) * DDIM + lane, 0, 0);
    float p = 0.f;
    for (int d = lane; d < DDIM; d += 32)
      p = fmaf(tanhf(wo_lds[d] + pv[d]), we_lds[d], p);
    for (int off = 16; off > 0; off >>= 1)
      p += __shfl_xor(p, off, 32);
    if (lane == 0) e_lds[l] = p;
  }
  __syncthreads();

  // --- softmax over L=512 in LDS ---
  float m = -3.402823466e+38f;
  for (int i = tid; i < LSEQ; i += 256) m = fmaxf(m, e_lds[i]);
  red[tid] = m; __syncthreads();
  for (int s = 128; s > 0; s >>= 1) {
    if (tid < s) red[tid] = fmaxf(red[tid], red[tid + s]);
    __syncthreads();
  }
  const float mx = red[0];
  __syncthreads();

  float sum = 0.f;
  for (int i = tid; i < LSEQ; i += 256) {
    const float v = __expf(e_lds[i] - mx);
    e_lds[i] = v;
    sum += v;
  }
  red[tid] = sum; __syncthreads();
  for (int s = 128; s > 0; s >>= 1) {
    if (tid < s) red[tid] += red[tid + s];
    __syncthreads();
  }
  const float inv = 1.f / red[0];

  float* aout = alpha_out + ((size_t)b * NCH + n) * LSEQ;
  for (int i = tid; i < LSEQ; i += 256) aout[i] = e_lds[i] * inv;
}

// ---------------------------------------------------------------------------
// Kernel 4: visual_features[b] = alpha[b](32x512) @ inp[b](512x512)  (WMMA)
// waves = 32 batches * 2 mtiles * 8 nstrips = 512 ; 8/block -> 64 blocks.
// ---------------------------------------------------------------------------
__global__ void vf_gemm_wmma_kernel(const float* __restrict__ alpha,
                                    const float* __restrict__ inp,
                                    float* __restrict__ vf) {
  const int lane = threadIdx.x & 31;
  const int wave = (blockIdx.x * blockDim.x + threadIdx.x) >> 5;
  const int b      = wave >> 4;                  // /16
  const int t      = wave & 15;
  const int row_base = (t >> 3) * 16;            // 0 or 16
  const int col_base = (t & 7) * 64;             // 0..448

  const float* A = alpha + (size_t)b * NCH * LSEQ;   // 32 x 512
  const float* B = inp   + (size_t)b * LSEQ * DDIM;  // 512 x 512
  float*       C = vf    + (size_t)b * NCH * DDIM;

  v8f c0 = {}, c1 = {}, c2 = {}, c3 = {};
  for (int k0 = 0; k0 < LSEQ; k0 += 4) {
    const v2f a  = ldA(A, LSEQ, row_base, k0, lane);
    const v2f b0 = ldB(B, DDIM, k0, col_base +  0, lane);
    const v2f b1 = ldB(B, DDIM, k0, col_base + 16, lane);
    const v2f b2 = ldB(B, DDIM, k0, col_base + 32, lane);
    const v2f b3 = ldB(B, DDIM, k0, col_base + 48, lane);
    c0 = WMMA_F32(a, b0, c0);
    c1 = WMMA_F32(a, b1, c1);
    c2 = WMMA_F32(a, b2, c2);
    c3 = WMMA_F32(a, b3, c3);
  }

  const int col  = col_base + (lane & 15);
  const int rsel = (lane >> 4) << 3;
  for (int v = 0; v < 8; ++v) {
    float* o = C + (size_t)(row_base + v + rsel) * DDIM + col;
    o[0] = c0[v]; o[16] = c1[v]; o[32] = c2[v]; o[48] = c3[v];
  }
}

// ---------------------------------------------------------------------------
// Kernel 5: logits = vf(1024x512) @ Wc_w(512x97) + Wc_b  (WMMA, N tail guard)
// waves = 64 mtiles * 7 ntiles = 448 ; 8/block -> 56 blocks.
// Guarded loads keep EXEC all-ones for the WMMA (selects, not branches).
// ---------------------------------------------------------------------------
__global__ void logits_gemm_wmma_kernel(const float* __restrict__ vf,
                                        const float* __restrict__ Wc_w,
                                        const float* __restrict__ Wc_b,
                                        float* __restrict__ logits) {
  const int lane = threadIdx.x & 31;
  const int wave = (blockIdx.x * blockDim.x + threadIdx.x) >> 5;
  const int NT = 7;                               // ceil(97/16)
  const int row_base = (wave / NT) * 16;
  const int col_base = (wave % NT) * 16;
  const int col = col_base + (lane & 15);
  const bool cok = (col < NDICT);

  v8f c = {};
  for (int k0 = 0; k0 < DDIM; k0 += 4) {
    const v2f a = ldA(vf, DDIM, row_base, k0, lane);
    const int k = k0 + ((lane >> 4) << 1);
    v2f bb;
    bb.x = cok ? Wc_w[(size_t)k * NDICT + col]       : 0.f;
    bb.y = cok ? Wc_w[(size_t)(k + 1) * NDICT + col] : 0.f;
    c = WMMA_F32(a, bb, c);
  }

  if (cok) {
    const float bias = Wc_b[col];
    const int rsel = (lane >> 4) << 3;
    for (int v = 0; v < 8; ++v)
      logits[(size_t)(row_base + v + rsel) * NDICT + col] = c[v] + bias;
  }
}

// ---------------------------------------------------------------------------
extern "C" void kernel_launch(void* const* d_in, const int* in_sizes, int n_in,
                              void* d_out, int out_size, void* d_ws, size_t ws_size,
                              hipStream_t stream) {
  (void)in_sizes; (void)n_in; (void)out_size; (void)ws_size;

  const float* inp  = (const float*)d_in[0];   // (32, 512, 512)
  const float* emb  = (const float*)d_in[1];   // (32, 512)
  const float* Wo_w = (const float*)d_in[2];   // (512, 512)
  const float* Wo_b = (const float*)d_in[3];   // (512,)
  const float* Wv_w = (const float*)d_in[4];   // (512, 512)
  const float* Wv_b = (const float*)d_in[5];   // (512,)
  const float* We_w = (const float*)d_in[6];   // (512, 1)
  // const float* We_b = (const float*)d_in[7]; // softmax-invariant shift
  const float* Wc_w = (const float*)d_in[8];   // (512, 97)
  const float* Wc_b = (const float*)d_in[9];   // (97,)

  float* ws_wo    = (float*)d_ws;                                 // 32*512
  float* ws_wv    = ws_wo + (size_t)NCH * DDIM;                   // 16384*512
  float* ws_alpha = ws_wv + (size_t)BSZ * LSEQ * DDIM;            // 32*32*512

  float* out_vf     = (float*)d_out;                              // (32,32,512)
  float* out_logits = out_vf + (size_t)BSZ * NCH * DDIM;          // (32,32,97)

  // 1) w_o = emb @ Wo_w + Wo_b
  wo_gemm_kernel<<<(NCH * DDIM) / 256, 256, 0, stream>>>(emb, Wo_w, Wo_b, ws_wo);

  // 2) w_v = inp @ Wv_w + Wv_b   (f32 WMMA)
  wv_gemm_wmma_kernel<<<1024, 256, 0, stream>>>(inp, Wv_w, Wv_b, ws_wv);

  // 3) scores + softmax -> alpha
  attn_softmax_kernel<<<BSZ * NCH, 256, 0, stream>>>(ws_wo, ws_wv, We_w, ws_alpha);

  // 4) visual_features = alpha @ inp   (batched f32 WMMA)
  vf_gemm_wmma_kernel<<<64, 256, 0, stream>>>(ws_alpha, inp, out_vf);

  // 5) logits = vf @ Wc_w + Wc_b      (f32 WMMA, N=97 tail)
  logits_gemm_wmma_kernel<<<56, 256, 0, stream>>>(out_vf, Wc_w, Wc_b, out_logits);
}